// DCConv3dKernelUnitPolynomials_55516747268826
// MI455X (gfx1250) — compile-verified
//
#include <hip/hip_runtime.h>

typedef float v4f __attribute__((ext_vector_type(4)));
typedef int   v4i __attribute__((ext_vector_type(4)));

// Address-space-qualified pointer types for the gfx1250 async-LDS builtin:
// arg0 = global (__device__, AS1) int4*, arg1 = LDS (__shared__, AS3) int4*.
typedef __attribute__((address_space(1))) v4i* gv4i_p;
typedef __attribute__((address_space(3))) v4i* lv4i_p;

#define INV2PI 0.15915494309189535f      // v_cos/v_sin take revolutions
#define C_EXP  (-0.2404491734814939f)    // -(log2 e)/6 : exp(-r/6) = exp2(r*C_EXP)

#if defined(__HIP_DEVICE_COMPILE__) && __has_builtin(__builtin_amdgcn_global_load_async_to_lds_b128)
#define HAVE_ASYNC_LDS 1
#else
#define HAVE_ASYNC_LDS 0
#endif

// 15 folded coefficient*normalization products, staged in d_ws by fold_coeffs_kernel.
struct Wc { float w0,w1,w2,w3,w4,w5,w6,w7,w8,w9,w10,w11,w12,w13,w14; };

// ---------------------------------------------------------------------------
// Fold runtime coeffs c[0..13] with all static normalization constants.
// BASIS order: (1,0,0)(2,0,0)(2,1,-1)(2,1,0)(2,1,1)(3,0,0)(3,1,-1)(3,1,0)
//              (3,1,1)(3,2,-2)(3,2,-1)(3,2,0)(3,2,1)(3,2,2)
// ---------------------------------------------------------------------------
__global__ void fold_coeffs_kernel(const float* __restrict__ c, float* __restrict__ w) {
  if (threadIdx.x == 0 && blockIdx.x == 0) {
    const float A00  = 0.28209479177387814f;   // 1/sqrt(4pi)
    const float A1x  = 0.4886025119029199f;    // sqrt(3/4pi)  (== sqrt2*sqrt(3/8pi))
    const float A20h = 0.31539156525252005f;   // 0.5*sqrt(5/4pi)
    const float B21  = 1.0925484305920792f;    // 3*sqrt2*sqrt(5/24pi)
    const float B22  = 0.5462742152960396f;    // 3*sqrt2*sqrt(5/96pi)
    const float K21  = 0.20412414523193154f;   // 1/sqrt(24)
    const float K30  = 0.12830005981991683f;   // 2/sqrt(243)
    const float K31  = 0.04536092116265144f;   // 1/sqrt(486)
    const float K32e = 0.009016009177039787f;  // (4/9)/sqrt(2430)
    w[0]  =  c[0]  * 2.0f * A00;               // * e^-r
    w[1]  =  c[1]  * 0.5f * A00;               // * e2*(2-r)
    w[2]  = -c[2]  * A1x * K21;                // * e2*r * s*sin(phi)
    w[3]  =  c[3]  * A1x * K21;                // * e2*r * ct
    w[4]  = -c[4]  * A1x * K21;                // * e2*r * s*cos(phi)
    w[5]  =  c[5]  * K30 * A00;                // * e3*(3-2r+(2/9)r^2)
    w[6]  = -c[6]  * A1x * K31;                // * e3*Q31 * s*sin(phi)
    w[7]  =  c[7]  * A1x * K31;                // * e3*Q31 * ct
    w[8]  = -c[8]  * A1x * K31;                // * e3*Q31 * s*cos(phi)
    w[9]  =  c[9]  * B22 * K32e;               // * e3*r^2 * s^2*sin(2phi)
    w[10] = -c[10] * B21 * K32e;               // * e3*r^2 * ct*s*sin(phi)
    float b11 = c[11] * A20h * K32e;           // * e3*r^2 * (3ct^2-1)
    w[11] =  3.0f * b11;
    w[12] = -b11;
    w[13] = -c[12] * B21 * K32e;               // * e3*r^2 * ct*s*cos(phi)
    w[14] =  c[13] * B22 * K32e;               // * e3*r^2 * s^2*cos(2phi)
    w[15] = 0.0f;
  }
}

// ---------------------------------------------------------------------------
// Per-point evaluation: ~5 transcendentals + ~40 FMA/MUL.
// ---------------------------------------------------------------------------
__device__ __forceinline__ float eval_one(float r, float th, float ph, const Wc& W) {
  float u   = __builtin_amdgcn_exp2f(r * C_EXP);  // exp(-r/6)
  float e3  = u * u;                              // exp(-r/3)
  float e2  = e3 * u;                             // exp(-r/2)
  float e1  = e2 * e2;                            // exp(-r)
  float ct  = __builtin_amdgcn_cosf(th * INV2PI); // cos(theta)
  float pn  = ph * INV2PI;
  float cp  = __builtin_amdgcn_cosf(pn);          // cos(phi)
  float sp  = __builtin_amdgcn_sinf(pn);          // sin(phi)
  float ct2 = ct * ct;
  float s2  = fmaxf(__builtin_fmaf(-ct, ct, 1.0f), 0.0f);
  float s   = __builtin_amdgcn_sqrtf(s2);         // |sin(theta)|
  float t   = sp * cp;
  float s2p = t + t;                              // sin(2phi)
  float c2p = __builtin_fmaf(-2.0f * sp, sp, 1.0f); // cos(2phi)

  // n=1
  float acc = W.w0 * e1;
  // n=2, l=0
  acc = __builtin_fmaf(W.w1, e2 * (2.0f - r), acc);
  // n=2, l=1
  float r21 = e2 * r;
  float a1  = __builtin_fmaf(W.w4, cp, W.w2 * sp);
  a1  = __builtin_fmaf(W.w3, ct, a1 * s);
  acc = __builtin_fmaf(r21, a1, acc);
  // n=3, l=0 : P30 = 3 - 2r + (2/9)r^2
  float p30 = __builtin_fmaf(r, 0.2222222222222222f, -2.0f);
  p30 = __builtin_fmaf(r, p30, 3.0f);
  acc = __builtin_fmaf(W.w5, e3 * p30, acc);
  // n=3, l=1 : Q31 = (8/3)r - (4/9)r^2
  float q31 = __builtin_fmaf(r, -0.4444444444444444f, 2.6666666666666665f);
  float r31 = e3 * (q31 * r);
  float b1  = __builtin_fmaf(W.w8, cp, W.w6 * sp);
  b1  = __builtin_fmaf(W.w7, ct, b1 * s);
  acc = __builtin_fmaf(r31, b1, acc);
  // n=3, l=2
  float r32 = e3 * (r * r);
  float g = __builtin_fmaf(W.w14, c2p, W.w9 * s2p);
  g = g * s2;
  float h = __builtin_fmaf(W.w13, cp, W.w10 * sp);
  g = __builtin_fmaf(h, ct * s, g);
  g = g + __builtin_fmaf(ct2, W.w11, W.w12);
  acc = __builtin_fmaf(r32, g, acc);
  return acc;
}

__device__ __forceinline__ void wait_async0() {
#if __has_builtin(__builtin_amdgcn_s_wait_asynccnt)
  __builtin_amdgcn_s_wait_asynccnt(0);
#else
  asm volatile("s_wait_asynccnt 0x0" ::: "memory");
#endif
}

// ---------------------------------------------------------------------------
// Streaming kernel: 4 points / thread.
// Full blocks: GLOBAL_LOAD_ASYNC_TO_LDS_B128 stages the 12 KB block tile into
// LDS (ASYNCcnt-tracked, decoupled from consumer), then ds_load_b128 reads.
// Partial/tail blocks (and non-async toolchains): direct NT b128 loads.
// ---------------------------------------------------------------------------
__global__ __launch_bounds__(256)
void hydrogen_eval_kernel(const float* __restrict__ pos, const float* __restrict__ wg,
                          float* __restrict__ out, int nvec, int n) {
  const int tid = threadIdx.x;
  const int i   = blockIdx.x * 256 + tid;

  Wc W;  // uniform -> scalar loads
  W.w0  = wg[0];  W.w1  = wg[1];  W.w2  = wg[2];  W.w3  = wg[3];  W.w4  = wg[4];
  W.w5  = wg[5];  W.w6  = wg[6];  W.w7  = wg[7];  W.w8  = wg[8];  W.w9  = wg[9];
  W.w10 = wg[10]; W.w11 = wg[11]; W.w12 = wg[12]; W.w13 = wg[13]; W.w14 = wg[14];

  const v4f* p4 = (const v4f*)pos;

#if HAVE_ASYNC_LDS
  __shared__ v4f tile4[768];                     // 256 threads * 3 float4 = 12 KB
  const bool full_block = (blockIdx.x + 1) * 256 <= nvec;
  if (full_block) {
    const v4f* g = p4 + (size_t)3 * 256 * (size_t)blockIdx.x;
    // warm GL2 16 tile-strides (192 KB) ahead (speculative; OOB silently dropped)
    __builtin_prefetch((const char*)g + 196608, 0, 3);
#pragma unroll
    for (int j = 0; j < 3; ++j) {
      int idx = tid + j * 256;                   // coalesced: 512B per wave per issue
      __builtin_amdgcn_global_load_async_to_lds_b128(
          (gv4i_p)(g + idx), (lv4i_p)(&tile4[idx]), 0, 0);
    }
    wait_async0();                               // my wave's copies landed in LDS
    __syncthreads();                             // everyone's copies visible
    v4f q0 = tile4[3 * tid + 0];
    v4f q1 = tile4[3 * tid + 1];
    v4f q2 = tile4[3 * tid + 2];
    v4f o;
    o.x = eval_one(q0.x, q0.y, q0.z, W);
    o.y = eval_one(q0.w, q1.x, q1.y, W);
    o.z = eval_one(q1.z, q1.w, q2.x, W);
    o.w = eval_one(q2.y, q2.z, q2.w, W);
    __builtin_nontemporal_store(o, (v4f*)out + i);
  } else if (i < nvec)
#else
  if (i < nvec)
#endif
  {
    size_t base = (size_t)3 * (size_t)i;
    __builtin_prefetch((const char*)(p4 + base) + 196608, 0, 3);
    v4f q0 = __builtin_nontemporal_load(p4 + base + 0);
    v4f q1 = __builtin_nontemporal_load(p4 + base + 1);
    v4f q2 = __builtin_nontemporal_load(p4 + base + 2);
    v4f o;
    o.x = eval_one(q0.x, q0.y, q0.z, W);
    o.y = eval_one(q0.w, q1.x, q1.y, W);
    o.z = eval_one(q1.z, q1.w, q2.x, W);
    o.w = eval_one(q2.y, q2.z, q2.w, W);
    __builtin_nontemporal_store(o, (v4f*)out + i);
  }

  // scalar tail for n not divisible by 4 (no-op for the reference shapes)
  if (i == nvec) {
    for (int k = nvec * 4; k < n; ++k) {
      out[k] = eval_one(pos[3 * k], pos[3 * k + 1], pos[3 * k + 2], W);
    }
  }
}

// ---------------------------------------------------------------------------
extern "C" void kernel_launch(void* const* d_in, const int* in_sizes, int n_in,
                              void* d_out, int out_size, void* d_ws, size_t ws_size,
                              hipStream_t stream) {
  const float* pos  = (const float*)d_in[0];   // (OutN, n, 3) flat
  const float* coef = (const float*)d_in[1];   // 14 floats
  float* out = (float*)d_out;
  float* w   = (float*)d_ws;                   // 16 floats of scratch

  int n     = in_sizes[0] / 3;                 // number of points
  int nvec  = n >> 2;
  int extra = (n & 3) ? 1 : 0;

  fold_coeffs_kernel<<<1, 32, 0, stream>>>(coef, w);

  int threads = nvec + extra;
  if (threads < 1) threads = 1;
  const int block = 256;
  int grid = (threads + block - 1) / block;
  hydrogen_eval_kernel<<<grid, block, 0, stream>>>(pos, w, out, nvec, n);
}